// ICGALayer_58205396795708
// MI455X (gfx1250) — compile-verified
//
#include <hip/hip_runtime.h>
#include <hip/hip_bf16.h>
#include <math.h>

// Problem constants (from reference setup_inputs)
#define BB 4
#define NN 2048
#define EE 32768
#define HID 256
#define NH 8
#define HD 32
#define DFF 512
#define NEGV -1000000000.0f

typedef __attribute__((ext_vector_type(2))) float v2f;
typedef __attribute__((ext_vector_type(8))) float v8f;

// ---------------- monotonic float <-> uint mapping for atomic max ----------
__device__ __forceinline__ unsigned f2mono(float f) {
    unsigned u = __float_as_uint(f);
    return (u & 0x80000000u) ? ~u : (u | 0x80000000u);
}
__device__ __forceinline__ float mono2f(unsigned u) {
    unsigned v = (u & 0x80000000u) ? (u & 0x7FFFFFFFu) : ~u;
    return __uint_as_float(v);
}

// ---------------- fp32 WMMA GEMM: C[M,NC] = A[M,KD] @ B[KD,NC] (+bias)(+relu)
// Compile-time KD/NC so all strides are immediates. One wave computes a
// 16x64 tile: 4 accumulators share each A fragment (V_WMMA_F32_16X16X4_F32).
// Requires M%16==0, NC%64==0, KD%4==0, tiles % wavesPerBlock == 0.
template <int KD, int NC>
__global__ void gemm_wmma_f32(const float* __restrict__ A,
                              const float* __restrict__ Bm,
                              const float* __restrict__ bias,
                              float* __restrict__ C,
                              int M, int doRelu)
{
    const int wavesPerBlock = blockDim.x >> 5;
    const int waveId = threadIdx.x >> 5;
    const int lane   = threadIdx.x & 31;
    const int tilesN = NC >> 6;              // 64-wide n tiles
    const int tilesM = M >> 4;
    const int tileId = blockIdx.x * wavesPerBlock + waveId;
    if (tileId >= tilesM * tilesN) return;   // wave-uniform: EXEC stays all-1s

    const int tm = tileId / tilesN;
    const int tn = tileId % tilesN;
    const int half = lane >> 4;              // selects K pair {0,1} vs {2,3}
    const int l16  = lane & 15;

    // Per-lane base pointers hoisted out of the K loop (constant strides).
    const float* Ab = A  + (size_t)(tm * 16 + l16) * KD + half * 2;
    const float* Bb = Bm + (size_t)(half * 2) * NC + tn * 64 + l16;

    v8f acc[4];
    #pragma unroll
    for (int s = 0; s < 4; ++s) acc[s] = (v8f){0.f,0.f,0.f,0.f,0.f,0.f,0.f,0.f};

    #pragma unroll 8
    for (int k = 0; k < KD; k += 4) {
        v2f a = *(const v2f*)(Ab + k);       // A frag: row = l16, K pair by half
        #pragma unroll
        for (int s = 0; s < 4; ++s) {
            v2f b;                           // B frag: col = s*16 + l16
            b.x = Bb[(size_t)k * NC + s * 16];
            b.y = Bb[(size_t)k * NC + NC + s * 16];
            acc[s] = __builtin_amdgcn_wmma_f32_16x16x4_f32(
                         false, a, false, b, (short)0, acc[s], false, false);
        }
    }

    float* Cb = C + (size_t)(tm * 16) * NC + tn * 64;
    #pragma unroll
    for (int s = 0; s < 4; ++s) {
        float bval = bias ? bias[tn * 64 + s * 16 + l16] : 0.f;
        #pragma unroll
        for (int v = 0; v < 8; ++v) {
            int m = v + half * 8;            // C/D layout: row = vgpr + 8*half
            float x = acc[s][v] + bval;
            if (doRelu) x = fmaxf(x, 0.f);
            Cb[(size_t)m * NC + s * 16 + l16] = x;
        }
    }
}

// ---------------- init scratch: segmax <- mono(-1e9), denom <- 0, agg <- 0 --
__global__ void init_kernel(unsigned* __restrict__ segmax,
                            float* __restrict__ denom,
                            float* __restrict__ agg,
                            int nSeg, int nAgg)
{
    for (int i = blockIdx.x * blockDim.x + threadIdx.x; i < nAgg;
         i += gridDim.x * blockDim.x) {
        agg[i] = 0.f;
        if (i < nSeg) {
            segmax[i] = f2mono(NEGV);
            denom[i]  = 0.f;
        }
    }
}

// ---------------- per-edge logits + segment max -----------------------------
__global__ void edge_logits_kernel(const float* __restrict__ Q,
                                   const float* __restrict__ K,
                                   const float* __restrict__ ef,
                                   const float* __restrict__ M_tilde,
                                   const float* __restrict__ W_eb,
                                   const float* __restrict__ beta_p,
                                   const int* __restrict__ edge_index,
                                   const int* __restrict__ n_edges,
                                   float* __restrict__ logits,
                                   unsigned* __restrict__ segmax)
{
    int idx = blockIdx.x * blockDim.x + threadIdx.x;
    if (idx >= BB * EE) return;
    int b = idx / EE, e = idx % EE;
    int src = edge_index[(size_t)b * 2 * EE + e];
    int dst = edge_index[(size_t)b * 2 * EE + EE + e];
    bool valid = e < n_edges[b];

    float beta_t = tanhf(beta_p[0]);
    float medge  = M_tilde[(size_t)b * NN * NN + (size_t)src * NN + dst];
    float ef0 = ef[((size_t)b * EE + e) * 2 + 0];
    float ef1 = ef[((size_t)b * EE + e) * 2 + 1];
    const float scale = 0.17677669529663687f;   // 32^-0.5

    const float* q  = Q + (size_t)(b * NN + dst) * HID;
    const float* kk = K + (size_t)(b * NN + src) * HID;

    #pragma unroll
    for (int h = 0; h < NH; ++h) {
        float dot = 0.f;
        #pragma unroll
        for (int d = 0; d < HD; ++d)
            dot += q[h * HD + d] * kk[h * HD + d];
        float logit = dot * scale
                    + ef0 * W_eb[0 * NH + h] + ef1 * W_eb[1 * NH + h]
                    + beta_t * medge;
        size_t li = ((size_t)b * EE + e) * NH + h;
        if (valid) {
            logits[li] = logit;
            atomicMax(&segmax[(size_t)(b * NN + dst) * NH + h], f2mono(logit));
        } else {
            logits[li] = -INFINITY;
        }
    }
}

// ---------------- ex = exp(logit - max) (in-place), denom scatter-add -------
__global__ void ex_denom_kernel(float* __restrict__ logits,  // becomes ex
                                const unsigned* __restrict__ segmax,
                                const int* __restrict__ edge_index,
                                const int* __restrict__ n_edges,
                                float* __restrict__ denom)
{
    int idx = blockIdx.x * blockDim.x + threadIdx.x;
    if (idx >= BB * EE * NH) return;
    int h = idx % NH;
    int e = (idx / NH) % EE;
    int b = idx / (NH * EE);
    if (e >= n_edges[b]) { logits[idx] = 0.f; return; }
    int dst = edge_index[(size_t)b * 2 * EE + EE + e];
    float maxv = mono2f(segmax[(size_t)(b * NN + dst) * NH + h]);
    float ex = expf(logits[idx] - maxv);
    logits[idx] = ex;
    atomicAdd(&denom[(size_t)(b * NN + dst) * NH + h], ex);
}

// ---------------- agg[dst] += alpha * V[src] --------------------------------
__global__ void aggregate_kernel(const float* __restrict__ ex,
                                 const float* __restrict__ denom,
                                 const float* __restrict__ V,
                                 const int* __restrict__ edge_index,
                                 const int* __restrict__ n_edges,
                                 float* __restrict__ agg)
{
    long long idx = (long long)blockIdx.x * blockDim.x + threadIdx.x;
    if (idx >= (long long)BB * EE * HID) return;
    int c = (int)(idx % HID);
    int e = (int)((idx / HID) % EE);
    int b = (int)(idx / ((long long)HID * EE));
    if (e >= n_edges[b]) return;
    int src = edge_index[(size_t)b * 2 * EE + e];
    int dst = edge_index[(size_t)b * 2 * EE + EE + e];
    int h = c >> 5;
    float al = ex[((size_t)b * EE + e) * NH + h];
    float dn = denom[(size_t)(b * NN + dst) * NH + h];
    float alpha = al / fmaxf(dn, 1e-6f);
    atomicAdd(&agg[(size_t)(b * NN + dst) * HID + c],
              alpha * V[(size_t)(b * NN + src) * HID + c]);
}

// ---------------- out = LayerNorm(x + y) * g + bias -------------------------
__global__ void add_ln_kernel(const float* __restrict__ x,
                              const float* __restrict__ y,
                              const float* __restrict__ g,
                              const float* __restrict__ bias,
                              float* __restrict__ out)
{
    __shared__ float red[HID];
    int row = blockIdx.x;
    int t = threadIdx.x;
    float v = x[(size_t)row * HID + t] + y[(size_t)row * HID + t];
    red[t] = v;
    __syncthreads();
    for (int s = HID / 2; s > 0; s >>= 1) {
        if (t < s) red[t] += red[t + s];
        __syncthreads();
    }
    float mu = red[0] * (1.0f / HID);
    __syncthreads();
    float d = v - mu;
    red[t] = d * d;
    __syncthreads();
    for (int s = HID / 2; s > 0; s >>= 1) {
        if (t < s) red[t] += red[t + s];
        __syncthreads();
    }
    float var = red[0] * (1.0f / HID);
    float r = rsqrtf(var + 1e-5f);
    out[(size_t)row * HID + t] = d * r * g[t] + bias[t];
}

// ===========================================================================
extern "C" void kernel_launch(void* const* d_in, const int* in_sizes, int n_in,
                              void* d_out, int out_size, void* d_ws, size_t ws_size,
                              hipStream_t stream)
{
    const float* h        = (const float*)d_in[0];
    const float* ef       = (const float*)d_in[1];
    const float* M_tilde  = (const float*)d_in[2];
    const float* W_q      = (const float*)d_in[3];
    const float* W_k      = (const float*)d_in[4];
    const float* W_v      = (const float*)d_in[5];
    const float* W_o      = (const float*)d_in[6];
    const float* b_o      = (const float*)d_in[7];
    const float* W_eb     = (const float*)d_in[8];
    const float* beta_p   = (const float*)d_in[9];
    const float* g1       = (const float*)d_in[10];
    const float* b1       = (const float*)d_in[11];
    const float* g2       = (const float*)d_in[12];
    const float* b2       = (const float*)d_in[13];
    const float* W_ff1    = (const float*)d_in[14];
    const float* b_ff1    = (const float*)d_in[15];
    const float* W_ff2    = (const float*)d_in[16];
    const float* b_ff2    = (const float*)d_in[17];
    const int*   edge_idx = (const int*)d_in[18];
    const int*   n_edges  = (const int*)d_in[19];
    float* out = (float*)d_out;

    const int M = BB * NN;                 // 8192 rows
    const size_t NODE = (size_t)M * HID;   // 2,097,152

    // ---- workspace carve-up (floats) ----
    float*    ws    = (float*)d_ws;
    float*    Q     = ws;                  ws += NODE;
    float*    Kb    = ws;                  ws += NODE;
    float*    V     = ws;                  ws += NODE;
    float*    logit = ws;                  ws += (size_t)BB * EE * NH;   // becomes ex
    unsigned* segmx = (unsigned*)ws;       ws += (size_t)BB * NN * NH;
    float*    denom = ws;                  ws += (size_t)BB * NN * NH;
    float*    agg   = ws;                  ws += NODE;
    float*    out1  = ws;                  ws += NODE;
    float*    h1    = ws;                  ws += NODE;
    float*    ff1   = ws;                  ws += (size_t)M * DFF;
    float*    ff2   = ws;                  ws += NODE;

    const int TPB = 256;                   // 8 waves / block
    const int WPB = TPB / 32;

    // tiles are 16 rows x 64 cols per wave
    const int gridQKV = ((M / 16) * (HID / 64)) / WPB;   // 2048/8 = 256
    const int gridFF1 = ((M / 16) * (DFF / 64)) / WPB;   // 4096/8 = 512

    // Q, K, V projections
    gemm_wmma_f32<HID, HID><<<gridQKV, TPB, 0, stream>>>(h, W_q, nullptr, Q,  M, 0);
    gemm_wmma_f32<HID, HID><<<gridQKV, TPB, 0, stream>>>(h, W_k, nullptr, Kb, M, 0);
    gemm_wmma_f32<HID, HID><<<gridQKV, TPB, 0, stream>>>(h, W_v, nullptr, V,  M, 0);

    // init segmax / denom / agg
    {
        int nSeg = BB * NN * NH;
        int nAgg = (int)NODE;
        init_kernel<<<(nAgg + TPB - 1) / TPB, TPB, 0, stream>>>(segmx, denom, agg, nSeg, nAgg);
    }

    // edge logits + segment max
    edge_logits_kernel<<<(BB * EE + TPB - 1) / TPB, TPB, 0, stream>>>(
        Q, Kb, ef, M_tilde, W_eb, beta_p, edge_idx, n_edges, logit, segmx);

    // exp / denom
    ex_denom_kernel<<<(BB * EE * NH + TPB - 1) / TPB, TPB, 0, stream>>>(
        logit, segmx, edge_idx, n_edges, denom);

    // weighted aggregation
    {
        long long tot = (long long)BB * EE * HID;
        int blocks = (int)((tot + TPB - 1) / TPB);
        aggregate_kernel<<<blocks, TPB, 0, stream>>>(logit, denom, V, edge_idx, n_edges, agg);
    }

    // out = agg @ W_o + b_o
    gemm_wmma_f32<HID, HID><<<gridQKV, TPB, 0, stream>>>(agg, W_o, b_o, out1, M, 0);

    // h1 = LN(h + out)
    add_ln_kernel<<<M, HID, 0, stream>>>(h, out1, g1, b1, h1);

    // ff = relu(h1 @ W_ff1 + b_ff1) @ W_ff2 + b_ff2
    gemm_wmma_f32<HID, DFF><<<gridFF1, TPB, 0, stream>>>(h1,  W_ff1, b_ff1, ff1, M, 1);
    gemm_wmma_f32<DFF, HID><<<gridQKV, TPB, 0, stream>>>(ff1, W_ff2, b_ff2, ff2, M, 0);

    // final: LN(h1 + ff) -> d_out
    add_ln_kernel<<<M, HID, 0, stream>>>(h1, ff2, g2, b2, out);
}